// RetrieveAndReadFramework_37151467110400
// MI455X (gfx1250) — compile-verified
//
#include <hip/hip_runtime.h>

typedef __attribute__((ext_vector_type(2))) float v2f;
typedef __attribute__((ext_vector_type(8))) float v8f;

#define DIM 128

// ---------------------------------------------------------------------------
// Degree / normalization kernels
// ---------------------------------------------------------------------------
__global__ __launch_bounds__(256) void deg_init(float* __restrict__ deg, int n) {
    int i = blockIdx.x * blockDim.x + threadIdx.x;
    if (i < n) deg[i] = 1.0f;  // self-loop contributes 1 to in-degree
}

__global__ __launch_bounds__(256) void deg_count(const long long* __restrict__ dst,
                                                 float* __restrict__ deg, int ne) {
    int i = blockIdx.x * blockDim.x + threadIdx.x;
    if (i < ne) atomicAdd(&deg[(size_t)dst[i]], 1.0f);
}

__global__ __launch_bounds__(256) void deg_rsqrt(float* __restrict__ deg, int n) {
    int i = blockIdx.x * blockDim.x + threadIdx.x;
    if (i < n) {
        float d = deg[i];
        deg[i] = (d > 0.0f) ? rsqrtf(d) : 0.0f;  // becomes dinv, in place
    }
}

// ---------------------------------------------------------------------------
// Dense transform: H = X * W   (X: [N,128], W: [128,128], H: [N,128])
// FP32 WMMA 16x16x4. One wave per 16x16 output tile; 8 waves/block cover the
// 8 column tiles of DIM=128 for one 16-row stripe.
//
// A (16x4, MxK) per ISA 7.12.2: lanes 0-15 -> M=lane, VGPR0=K0/K2, VGPR1=K1/K3
//   => lane holds A[m][k0+kk], A[m][k0+kk+1] with m=lane&15, kk=(lane>>4)*2
// B (4x16, KxN): row K striped across lanes within a VGPR, mirrored layout:
//   => lane holds B[k0+kk][n0+n], B[k0+kk+1][n0+n] with n=lane&15
// C/D (16x16 f32): VGPR r -> rows r (lanes 0-15) and r+8 (lanes 16-31)
// ---------------------------------------------------------------------------
__global__ __launch_bounds__(256) void gcn_gemm_wmma(const float* __restrict__ X,
                                                     const float* __restrict__ W,
                                                     float* __restrict__ H,
                                                     int n_rows) {
    const int wave = threadIdx.x >> 5;   // 0..7 -> N tile
    const int lane = threadIdx.x & 31;
    const int m0   = blockIdx.x * 16;
    const int n0   = wave * 16;

    const int mn = lane & 15;            // M index for A, N index for B
    const int kk = (lane >> 4) << 1;     // 0 or 2

    int mrow = m0 + mn;
    if (mrow >= n_rows) mrow = n_rows - 1;   // clamp: keep EXEC all-ones for WMMA
    const float* __restrict__ xrow = X + (size_t)mrow * DIM;

    v8f c = {0.f, 0.f, 0.f, 0.f, 0.f, 0.f, 0.f, 0.f};

#pragma unroll
    for (int k0 = 0; k0 < DIM; k0 += 4) {
        v2f a, b;
        a.x = xrow[k0 + kk];
        a.y = xrow[k0 + kk + 1];
        b.x = W[(size_t)(k0 + kk) * DIM + n0 + mn];
        b.y = W[(size_t)(k0 + kk + 1) * DIM + n0 + mn];
        c = __builtin_amdgcn_wmma_f32_16x16x4_f32(
                /*neg_a=*/false, a, /*neg_b=*/false, b,
                /*c_mod=*/(short)0, c, /*reuse_a=*/false, /*reuse_b=*/false);
    }

    const int rowbase = (lane < 16) ? 0 : 8;
    const int col = lane & 15;
#pragma unroll
    for (int r = 0; r < 8; ++r) {
        int orow = m0 + rowbase + r;
        if (orow < n_rows) H[(size_t)orow * DIM + n0 + col] = c[r];
    }
}

// ---------------------------------------------------------------------------
// acc = h * dinv^2  (self-loop term; also initializes the accumulator)
// ---------------------------------------------------------------------------
__global__ __launch_bounds__(256) void self_loop_init(const float* __restrict__ h,
                                                      const float* __restrict__ dinv,
                                                      float* __restrict__ acc, int n) {
    int i = blockIdx.x * blockDim.x + threadIdx.x;
    if (i < n) {
        float di = dinv[i >> 7];   // node = i / 128
        acc[i] = h[i] * di * di;
    }
}

// ---------------------------------------------------------------------------
// Edge scatter: acc[dst] += h[src] * dinv[src]*dinv[dst]
// 32 lanes per edge; each lane handles 4 columns via float4 gather + atomics.
// ---------------------------------------------------------------------------
__global__ __launch_bounds__(256) void edge_scatter(const long long* __restrict__ src,
                                                    const long long* __restrict__ dst,
                                                    const float* __restrict__ dinv,
                                                    const float* __restrict__ h,
                                                    float* __restrict__ acc, int ne) {
    int t = blockIdx.x * blockDim.x + threadIdx.x;
    int e = t >> 5;
    if (e >= ne) return;
    int lane = t & 31;

    long long s = src[e];
    long long d = dst[e];
    float norm = dinv[(size_t)s] * dinv[(size_t)d];

    const float4* hp = (const float4*)(h + (size_t)s * DIM);
    float4 v = hp[lane];
    float* op = acc + (size_t)d * DIM + lane * 4;
    atomicAdd(op + 0, v.x * norm);
    atomicAdd(op + 1, v.y * norm);
    atomicAdd(op + 2, v.z * norm);
    atomicAdd(op + 3, v.w * norm);
}

// ---------------------------------------------------------------------------
// xout = relu(acc + bias)
// ---------------------------------------------------------------------------
__global__ __launch_bounds__(256) void bias_relu(const float* __restrict__ acc,
                                                 const float* __restrict__ bias,
                                                 float* __restrict__ xout, int n) {
    int i = blockIdx.x * blockDim.x + threadIdx.x;
    if (i < n) {
        float v = acc[i] + bias[i & (DIM - 1)];
        xout[i] = fmaxf(v, 0.0f);
    }
}

// ---------------------------------------------------------------------------
extern "C" void kernel_launch(void* const* d_in, const int* in_sizes, int n_in,
                              void* d_out, int out_size, void* d_ws, size_t ws_size,
                              hipStream_t stream) {
    const float*     x0 = (const float*)d_in[0];
    const long long* ei = (const long long*)d_in[1];   // int64 edge_index [2, E]
    const float*     Wl = (const float*)d_in[2];       // [L, 128, 128]
    const float*     Bl = (const float*)d_in[3];       // [L, 128]
    float*           out = (float*)d_out;

    const int N = in_sizes[0] / DIM;
    const int E = in_sizes[1] / 2;
    const int L = in_sizes[2] / (DIM * DIM);

    const long long* src = ei;
    const long long* dst = ei + E;

    const size_t nd = (size_t)N * DIM;

    float* ws   = (float*)d_ws;
    float* dinv = ws;                                  // N floats
    float* H    = ws + (((size_t)N + 255) & ~(size_t)255);
    float* P0   = H + nd;
    float* P1   = P0 + nd;

    // --- normalization: dinv = rsqrt(in_degree + 1) ---
    deg_init <<<(N + 255) / 256, 256, 0, stream>>>(dinv, N);
    deg_count<<<(E + 255) / 256, 256, 0, stream>>>(dst, dinv, E);
    deg_rsqrt<<<(N + 255) / 256, 256, 0, stream>>>(dinv, N);

    const int nGemmBlocks = (N + 15) / 16;
    const int nElemBlocks = (int)((nd + 255) / 256);
    const int nEdgeBlocks = (int)(((size_t)E * 32 + 255) / 256);

    const float* Xcur = x0;
    float* Pb = P0;
    float* Pother = P1;

    for (int l = 0; l < L; ++l) {
        const float* W = Wl + (size_t)l * DIM * DIM;
        const float* b = Bl + (size_t)l * DIM;

        gcn_gemm_wmma <<<nGemmBlocks, 256, 0, stream>>>(Xcur, W, H, N);
        self_loop_init<<<nElemBlocks, 256, 0, stream>>>(H, dinv, Pb, (int)nd);
        edge_scatter  <<<nEdgeBlocks, 256, 0, stream>>>(src, dst, dinv, H, Pb, E);

        float* xnext = (l == L - 1) ? out : Pb;        // in-place except last layer
        bias_relu     <<<nElemBlocks, 256, 0, stream>>>(Pb, b, xnext, (int)nd);

        Xcur = xnext;
        float* t = Pb; Pb = Pother; Pother = t;        // fresh accumulator next layer
    }
}